// Model_78262894068348
// MI455X (gfx1250) — compile-verified
//
#include <hip/hip_runtime.h>
#include <hip/hip_bf16.h>

#define D_FEAT 128
#define NEG_SLOPE 0.2f

typedef __attribute__((ext_vector_type(2))) float v2f;
typedef __attribute__((ext_vector_type(8))) float v8f;

// ---------------------------------------------------------------------------
// K1: feat = x @ W via V_WMMA_F32_16X16X4_F32. One wave computes one 16x16
// output tile. A: 16x4 f32 (2 VGPR/lane), B: 4x16 f32 (2 VGPR/lane),
// C/D: 16x16 f32 accumulator (8 VGPR/lane). K-loop: 128/4 = 32 WMMA issues.
// Layouts per CDNA5 ISA 7.12.2:
//   A lane L<16 : row M=L,  v0=K, v1=K+1 ; lane L>=16: row M=L-16, v0=K+2, v1=K+3
//   B lane L<16 : col N=L,  v0=K, v1=K+1 ; lane L>=16: col N=L-16, v0=K+2, v1=K+3
//   D lane L<16 : vj -> (M=j,   N=L)    ; lane L>=16: vj -> (M=j+8, N=L-16)
// ---------------------------------------------------------------------------
__global__ __launch_bounds__(256) void gat_gemm_wmma(
    const float* __restrict__ x, const float* __restrict__ W,
    float* __restrict__ feat, int M)
{
    const int wave = (blockIdx.x * blockDim.x + threadIdx.x) >> 5;
    const int lane = threadIdx.x & 31;
    const int ntiles = D_FEAT / 16;                 // 8
    const int mt = wave / ntiles;
    const int nt = wave % ntiles;
    const int Mt = (M + 15) / 16;
    if (mt >= Mt) return;                           // uniform per wave

    const int half = lane >> 4;                     // 0 | 1
    const int l16  = lane & 15;
    const int arow = mt * 16 + l16;                 // A row for this lane
    const int bcol = nt * 16 + l16;                 // B/D column for this lane

    v8f acc = {};
    #pragma unroll 4
    for (int k = 0; k < D_FEAT; k += 4) {
        const int ka = k + 2 * half;                // even -> 8B aligned
        v2f a = *(const v2f*)(x + (size_t)arow * D_FEAT + ka);
        v2f b;
        b.x = W[(size_t)ka       * D_FEAT + bcol];
        b.y = W[(size_t)(ka + 1) * D_FEAT + bcol];
        acc = __builtin_amdgcn_wmma_f32_16x16x4_f32(
            /*neg_a=*/false, a, /*neg_b=*/false, b,
            /*c_mod=*/(short)0, acc, /*reuse_a=*/false, /*reuse_b=*/false);
    }

    #pragma unroll
    for (int j = 0; j < 8; ++j) {
        const int r = mt * 16 + j + 8 * half;
        feat[(size_t)r * D_FEAT + bcol] = acc[j];
    }
}

// ---------------------------------------------------------------------------
// K2: el[n] = feat[n,:].attn_l ; er[n] = feat[n,:].attn_r. One wave per node.
// ---------------------------------------------------------------------------
__global__ __launch_bounds__(256) void attn_proj(
    const float* __restrict__ feat, const float* __restrict__ al,
    const float* __restrict__ ar, float* __restrict__ el,
    float* __restrict__ er, int M)
{
    const int wave = (blockIdx.x * blockDim.x + threadIdx.x) >> 5;
    const int lane = threadIdx.x & 31;
    if (wave >= M) return;
    float sl = 0.f, sr = 0.f;
    #pragma unroll
    for (int j = lane; j < D_FEAT; j += 32) {
        const float f = feat[(size_t)wave * D_FEAT + j];
        sl += f * al[j];
        sr += f * ar[j];
    }
    #pragma unroll
    for (int off = 16; off > 0; off >>= 1) {
        sl += __shfl_xor(sl, off, 32);
        sr += __shfl_xor(sr, off, 32);
    }
    if (lane == 0) { el[wave] = sl; er[wave] = sr; }
}

// order-preserving float <-> uint key (exact float max via integer atomicMax)
__device__ __forceinline__ unsigned f2key(float f) {
    unsigned u = __float_as_uint(f);
    return (u & 0x80000000u) ? ~u : (u | 0x80000000u);
}
__device__ __forceinline__ float key2f(unsigned k) {
    unsigned u = (k & 0x80000000u) ? (k & 0x7FFFFFFFu) : ~k;
    return __uint_as_float(u);
}

// ---------------------------------------------------------------------------
// K3: per-edge leaky_relu logit; segment-max into mkey[dst] (uint atomicMax).
// ---------------------------------------------------------------------------
__global__ __launch_bounds__(256) void edge_max(
    const float* __restrict__ el, const float* __restrict__ er,
    const int* __restrict__ src, const int* __restrict__ dst,
    unsigned* __restrict__ mkey, int E)
{
    const int i = blockIdx.x * blockDim.x + threadIdx.x;
    if (i >= E) return;
    float e = el[src[i]] + er[dst[i]];
    e = e > 0.f ? e : NEG_SLOPE * e;
    atomicMax(&mkey[dst[i]], f2key(e));
}

// ---------------------------------------------------------------------------
// K4: one wave per edge. ee = exp(e - m[dst]); s[dst] += ee;
//     h[dst,:] += ee * feat[src,:]  (un-normalized; divide by s later).
// Gathers/atomics hit L2 (feat/h are 25.6 MB each, L2 is 192 MB).
// ---------------------------------------------------------------------------
__global__ __launch_bounds__(256) void edge_accum(
    const float* __restrict__ el, const float* __restrict__ er,
    const float* __restrict__ feat, const int* __restrict__ src,
    const int* __restrict__ dst, const unsigned* __restrict__ mkey,
    float* __restrict__ s, float* __restrict__ h, int E)
{
    const int wave = (blockIdx.x * blockDim.x + threadIdx.x) >> 5;
    const int lane = threadIdx.x & 31;
    if (wave >= E) return;
    const int sn = src[wave];
    const int dn = dst[wave];
    float e = el[sn] + er[dn];
    e = e > 0.f ? e : NEG_SLOPE * e;
    const float m  = key2f(mkey[dn]);
    const float ee = __expf(e - m);
    if (lane == 0) atomicAdd(&s[dn], ee);
    #pragma unroll
    for (int j = lane; j < D_FEAT; j += 32)
        atomicAdd(&h[(size_t)dn * D_FEAT + j], ee * feat[(size_t)sn * D_FEAT + j]);
}

// ---------------------------------------------------------------------------
// K5: h = relu(h / s + bias). One thread per (node, feature).
// ---------------------------------------------------------------------------
__global__ __launch_bounds__(256) void node_finalize(
    const float* __restrict__ s, const float* __restrict__ bias,
    float* __restrict__ h, int M)
{
    const int i = blockIdx.x * blockDim.x + threadIdx.x;
    if (i >= M * D_FEAT) return;
    const int n = i >> 7;          // / 128
    const int j = i & 127;
    const float sv = s[n];
    float v = (sv > 0.f ? h[i] / sv : 0.f) + bias[j];
    h[i] = v > 0.f ? v : 0.f;
}

// ---------------------------------------------------------------------------
// K6: score[i] = dot(h[a[i],:], h[b[i],:]). One wave per pair.
// ---------------------------------------------------------------------------
__global__ __launch_bounds__(256) void pair_score(
    const float* __restrict__ h, const int* __restrict__ a,
    const int* __restrict__ b, float* __restrict__ out, int P)
{
    const int wave = (blockIdx.x * blockDim.x + threadIdx.x) >> 5;
    const int lane = threadIdx.x & 31;
    if (wave >= P) return;
    const int u = a[wave];
    const int v = b[wave];
    float d = 0.f;
    #pragma unroll
    for (int j = lane; j < D_FEAT; j += 32)
        d += h[(size_t)u * D_FEAT + j] * h[(size_t)v * D_FEAT + j];
    #pragma unroll
    for (int off = 16; off > 0; off >>= 1)
        d += __shfl_xor(d, off, 32);
    if (lane == 0) out[wave] = d;
}

extern "C" void kernel_launch(void* const* d_in, const int* in_sizes, int n_in,
                              void* d_out, int out_size, void* d_ws, size_t ws_size,
                              hipStream_t stream) {
    const float* x    = (const float*)d_in[0];
    const float* W    = (const float*)d_in[1];
    const float* al   = (const float*)d_in[2];
    const float* ar   = (const float*)d_in[3];
    const float* bias = (const float*)d_in[4];
    const int* src  = (const int*)d_in[5];
    const int* dst  = (const int*)d_in[6];
    const int* psrc = (const int*)d_in[7];
    const int* pdst = (const int*)d_in[8];
    const int* nsrc = (const int*)d_in[9];
    const int* ndst = (const int*)d_in[10];
    float* out = (float*)d_out;

    const int M = in_sizes[0] / D_FEAT;   // 50000
    const int E = in_sizes[5];            // 800000
    const int P = in_sizes[7];            // 200000
    const int Q = in_sizes[9];            // 200000

    // workspace carve (all 256B aligned since sizes are multiples of 256B-ish)
    char* p = (char*)d_ws;
    float* feat = (float*)p; p += (size_t)M * D_FEAT * sizeof(float);
    float* h    = (float*)p; p += (size_t)M * D_FEAT * sizeof(float);
    float* el   = (float*)p; p += (size_t)M * sizeof(float);
    float* er   = (float*)p; p += (size_t)M * sizeof(float);
    float* s    = (float*)p; p += (size_t)M * sizeof(float);
    unsigned* mkey = (unsigned*)p;

    // zero accumulators each call (capture-safe async memsets)
    hipMemsetAsync(h, 0, (size_t)M * D_FEAT * sizeof(float), stream);
    hipMemsetAsync(s, 0, (size_t)M * sizeof(float), stream);
    hipMemsetAsync(mkey, 0, (size_t)M * sizeof(unsigned), stream);

    const int TPB = 256;                  // 8 wave32 per block

    // K1: GEMM — one wave per 16x16 tile
    {
        const int waves = ((M + 15) / 16) * (D_FEAT / 16);
        const int blocks = (waves * 32 + TPB - 1) / TPB;
        gat_gemm_wmma<<<blocks, TPB, 0, stream>>>(x, W, feat, M);
    }
    // K2: attention projections — one wave per node
    {
        const int blocks = (M * 32 + TPB - 1) / TPB;
        attn_proj<<<blocks, TPB, 0, stream>>>(feat, al, ar, el, er, M);
    }
    // K3: segment max — one thread per edge
    {
        const int blocks = (E + TPB - 1) / TPB;
        edge_max<<<blocks, TPB, 0, stream>>>(el, er, src, dst, mkey, E);
    }
    // K4: softmax-numerator scatter — one wave per edge
    {
        const long long threads = (long long)E * 32;
        const int blocks = (int)((threads + TPB - 1) / TPB);
        edge_accum<<<blocks, TPB, 0, stream>>>(el, er, feat, src, dst, mkey, s, h, E);
    }
    // K5: normalize + bias + relu — one thread per (node, feature)
    {
        const int blocks = (M * D_FEAT + TPB - 1) / TPB;
        node_finalize<<<blocks, TPB, 0, stream>>>(s, bias, h, M);
    }
    // K6: pos / neg scores — one wave per pair
    {
        const int blocks_p = (P * 32 + TPB - 1) / TPB;
        pair_score<<<blocks_p, TPB, 0, stream>>>(h, psrc, pdst, out, P);
        const int blocks_n = (Q * 32 + TPB - 1) / TPB;
        pair_score<<<blocks_n, TPB, 0, stream>>>(h, nsrc, ndst, out + P, Q);
    }
}